// GraphAttentionLayer_75376676045018
// MI455X (gfx1250) — compile-verified
//
#include <hip/hip_runtime.h>
#include <hip/hip_bf16.h>
#include <math.h>

// ---------------------------------------------------------------------------
// GAT layer, MI455X (gfx1250, wave32).
// Roofline: adj = 256 MB is the only large operand -> ~11us floor @ 23.3 TB/s.
// Strategy: never materialize the 8192x8192 score/attention matrices.
//   K1: Wh = x @ W^T with V_WMMA_F32_16X16X4_F32 (fp32 WMMA path).
//   K2: e = Wh@a1, f = Wh@a2 (tiny).
//   K3: one workgroup per row: stream the adj row once (non-temporal b128
//       loads), record nonzero (j, leakyrelu(e_i+f_j)) in an LDS list
//       (avg degree ~33), LDS-reduce max & denom, then gather only the
//       listed Wh rows (Wh = 4MB, L2-resident) for the weighted sum + ELU.
// ---------------------------------------------------------------------------

typedef __attribute__((ext_vector_type(2))) float v2f;
typedef __attribute__((ext_vector_type(4))) float v4f;
typedef __attribute__((ext_vector_type(8))) float v8f;

#define N_NODES 8192
#define IN_F    256
#define OUT_F   128
#define ALPHA   0.2f
#define LIST_CAP 2048   // avg nonzeros/row ~33 (p=0.004); 2048 is unreachable

// ---------------------------------------------------------------------------
// Kernel 1: Wh = x @ W^T via V_WMMA_F32_16X16X4_F32.
// One wave computes one 16x16 output tile; K-loop of 64 WMMAs over K=256.
// A (16x4 f32): lanes 0-15 = M rows, VGPR pair = K{0,1}; lanes 16-31 = K{2,3}.
// B (4x16 f32): mirrored with N on lanes. C/D: 8 VGPRs, lanes 0-15 M=r,
// lanes 16-31 M=r+8 (ISA 7.12.2 layout).
// ---------------------------------------------------------------------------
__global__ __launch_bounds__(128) void gat_wh_wmma(
    const float* __restrict__ x,   // (N, IN_F)
    const float* __restrict__ W,   // (OUT_F, IN_F)
    float* __restrict__ Wh)        // (N, OUT_F)
{
    const int lane  = threadIdx.x & 31;
    const int wave  = threadIdx.x >> 5;
    const int tile  = blockIdx.x * 4 + wave;          // 4096 tiles total
    const int tpr   = OUT_F / 16;                     // 8 col-tiles per row-band
    const int m0    = (tile / tpr) * 16;
    const int n0    = (tile % tpr) * 16;

    const int lmod  = lane & 15;                      // M (for A) / N (for B)
    const int lhalf = lane >> 4;                      // selects K pair {0,1}/{2,3}

    const float* __restrict__ xrow = x + (size_t)(m0 + lmod) * IN_F + 2 * lhalf;
    const float* __restrict__ wrow = W + (size_t)(n0 + lmod) * IN_F + 2 * lhalf;

    v8f c = {0.f, 0.f, 0.f, 0.f, 0.f, 0.f, 0.f, 0.f};

    #pragma unroll 4
    for (int k = 0; k < IN_F; k += 4) {
        v2f a = *(const v2f*)(xrow + k);   // x[m0+lmod, k + 2*lhalf .. +1]
        v2f b = *(const v2f*)(wrow + k);   // W[n0+lmod, k + 2*lhalf .. +1]
        // 8 args: (neg_a, A, neg_b, B, c_mod, C, reuse_a, reuse_b)
        c = __builtin_amdgcn_wmma_f32_16x16x4_f32(
                false, a, false, b, (short)0, c, false, false);
    }

    // Store D per 32-bit 16x16 C/D layout.
    float* __restrict__ out = Wh + (size_t)(m0 + lhalf * 8) * OUT_F + n0 + lmod;
    #pragma unroll
    for (int r = 0; r < 8; ++r)
        out[(size_t)r * OUT_F] = c[r];
}

// ---------------------------------------------------------------------------
// Kernel 2: e[i] = dot(Wh[i], a[:128]), f[i] = dot(Wh[i], a[128:]).
// One 128-thread block per row; LDS tree reduce.
// ---------------------------------------------------------------------------
__global__ __launch_bounds__(128) void gat_ef(
    const float* __restrict__ Wh,
    const float* __restrict__ a,
    float* __restrict__ e,
    float* __restrict__ f)
{
    __shared__ float se[128];
    __shared__ float sf[128];
    const int i = blockIdx.x;
    const int t = threadIdx.x;

    const float wh = Wh[(size_t)i * OUT_F + t];
    se[t] = wh * a[t];
    sf[t] = wh * a[OUT_F + t];
    __syncthreads();
    #pragma unroll
    for (int s = 64; s > 0; s >>= 1) {
        if (t < s) { se[t] += se[t + s]; sf[t] += sf[t + s]; }
        __syncthreads();
    }
    if (t == 0) { e[i] = se[0]; f[i] = sf[0]; }
}

// ---------------------------------------------------------------------------
// Kernel 3: fused mask + softmax + sparse aggregation + ELU. One 256-thread
// block (8 waves) per row. Single non-temporal streaming pass over the adj
// row (32 KB as b128 loads); nonzeros go to an LDS (j, score) list.
// ---------------------------------------------------------------------------
__global__ __launch_bounds__(256) void gat_attn(
    const float* __restrict__ adj,   // (N, N)
    const float* __restrict__ Wh,    // (N, OUT_F)  -- L2 resident (4 MB)
    const float* __restrict__ e,
    const float* __restrict__ f,
    float* __restrict__ out)         // (N, OUT_F)
{
    __shared__ int      s_idx[LIST_CAP];
    __shared__ float    s_sc[LIST_CAP];
    __shared__ unsigned s_cnt;
    __shared__ float    s_red[256];
    __shared__ float    s_acc[OUT_F];

    const int i = blockIdx.x;
    const int t = threadIdx.x;
    if (t == 0) s_cnt = 0u;
    __syncthreads();

    const float ei = e[i];
    const v4f* __restrict__ adjrow = (const v4f*)(adj + (size_t)i * N_NODES);

    // Phase 1: stream adj row (NT hint: no reuse), build candidate list.
    float lmax = -3.0e38f;
    for (int v = t; v < N_NODES / 4; v += 256) {           // 8 iters/thread
        v4f av = __builtin_nontemporal_load(&adjrow[v]);   // global_load_b128 NT
        const int jbase = v * 4;
        #pragma unroll
        for (int q = 0; q < 4; ++q) {
            if (av[q] > 0.f) {
                const int j = jbase + q;
                float s = ei + f[j];
                s = (s > 0.f) ? s : ALPHA * s;             // leaky_relu
                const unsigned p = atomicAdd(&s_cnt, 1u);  // ds_add_rtn_u32
                if (p < LIST_CAP) { s_idx[p] = j; s_sc[p] = s; }
                lmax = fmaxf(lmax, s);
            }
        }
    }

    // Phase 2: block max reduce.
    s_red[t] = lmax;
    __syncthreads();
    #pragma unroll
    for (int s = 128; s > 0; s >>= 1) {
        if (t < s) s_red[t] = fmaxf(s_red[t], s_red[t + s]);
        __syncthreads();
    }
    const float M = s_red[0];
    __syncthreads();

    const unsigned c0 = s_cnt;
    const int cnt = (c0 < (unsigned)LIST_CAP) ? (int)c0 : LIST_CAP;

    // Phase 3: scores -> exp(s - M) in place; partial denominator.
    float dsum = 0.f;
    for (int p = t; p < cnt; p += 256) {
        const float w = __expf(s_sc[p] - M);
        s_sc[p] = w;
        dsum += w;
    }
    __syncthreads();
    s_red[t] = dsum;
    __syncthreads();
    #pragma unroll
    for (int s = 128; s > 0; s >>= 1) {
        if (t < s) s_red[t] += s_red[t + s];
        __syncthreads();
    }
    const float invD = 1.0f / s_red[0];

    // Phase 4: sparse gather-accumulate. Feature k = t&127; the two halves of
    // the block split the list by parity, combined through LDS.
    const int k    = t & 127;
    const int half = t >> 7;
    float acc = 0.f;
    for (int p = half; p < cnt; p += 2)
        acc += s_sc[p] * Wh[(size_t)s_idx[p] * OUT_F + k];   // coalesced, L2 hit

    if (half == 1) s_acc[k] = acc;
    __syncthreads();
    if (half == 0) {
        const float h = (acc + s_acc[k]) * invD;
        out[(size_t)i * OUT_F + k] = (h > 0.f) ? h : (__expf(h) - 1.0f);  // ELU
    }
}

// ---------------------------------------------------------------------------
// Launch. Inputs: x (N*IN_F), adj (N*N), W (OUT_F*IN_F), a (2*OUT_F).
// Workspace: Wh (4 MB) + e + f.
// ---------------------------------------------------------------------------
extern "C" void kernel_launch(void* const* d_in, const int* in_sizes, int n_in,
                              void* d_out, int out_size, void* d_ws, size_t ws_size,
                              hipStream_t stream) {
    const float* x   = (const float*)d_in[0];
    const float* adj = (const float*)d_in[1];
    const float* W   = (const float*)d_in[2];
    const float* a   = (const float*)d_in[3];
    float* out = (float*)d_out;

    float* Wh = (float*)d_ws;                       // N * OUT_F
    float* e  = Wh + (size_t)N_NODES * OUT_F;       // N
    float* f  = e + N_NODES;                        // N

    // K1: 4096 16x16 tiles, 4 waves (tiles) per 128-thread block.
    gat_wh_wmma<<<(N_NODES / 16) * (OUT_F / 16) / 4, 128, 0, stream>>>(x, W, Wh);
    // K2: one block per row.
    gat_ef<<<N_NODES, 128, 0, stream>>>(Wh, a, e, f);
    // K3: one block per row, streams adj exactly once.
    gat_attn<<<N_NODES, 256, 0, stream>>>(adj, Wh, e, f, out);
}